// RNN_77060303225291
// MI455X (gfx1250) — compile-verified
//
#include <hip/hip_runtime.h>
#include <hip/hip_bf16.h>
#include <cmath>

typedef __attribute__((ext_vector_type(2))) float v2f;
typedef __attribute__((ext_vector_type(8))) float v8f;
typedef __attribute__((ext_vector_type(2))) int   v2i;

#define HID   1024
#define BATCH 8
#define SEQ   256
#define VOCAB 32000

#define AS1 __attribute__((address_space(1)))
#define AS3 __attribute__((address_space(3)))

// ---------------------------------------------------------------------------
// Async global->LDS copy (ASYNCcnt-tracked). The builtin takes pointers to
// 2 x i32 vectors (b64 payload): (global src, lds dst, imm offset, imm cpol).
// ---------------------------------------------------------------------------
#if __has_builtin(__builtin_amdgcn_global_load_async_to_lds_b64)
#define HAVE_ASYNC_LDS 1
__device__ __forceinline__ void async_cp_b64(const float* g, float* l) {
    __builtin_amdgcn_global_load_async_to_lds_b64(
        (AS1 v2i*)g, (AS3 v2i*)l, 0, 0);
}
__device__ __forceinline__ void async_wait_all() {
#if __has_builtin(__builtin_amdgcn_s_wait_asynccnt)
    __builtin_amdgcn_s_wait_asynccnt(0);
#else
    asm volatile("s_wait_asynccnt 0x0" ::: "memory");
#endif
}
#else
#define HAVE_ASYNC_LDS 0
__device__ __forceinline__ void async_cp_b64(const float* g, float* l) {
    *(v2f*)l = *(const v2f*)g;
}
__device__ __forceinline__ void async_wait_all() {}
#endif

__device__ __forceinline__ v8f wmma4(v2f a, v2f b, v8f c) {
    return __builtin_amdgcn_wmma_f32_16x16x4_f32(false, a, false, b,
                                                 (short)0, c, false, false);
}

// ---------------------------------------------------------------------------
// 1) Embedding gather: emb[t*8+b][:] = embed[x[b][t]][:]   ([T][B][H] layout)
// ---------------------------------------------------------------------------
__global__ __launch_bounds__(256)
void emb_gather_kernel(const int* __restrict__ x,
                       const float* __restrict__ embed,
                       float* __restrict__ emb) {
    int r = blockIdx.x;             // r = t*8 + b
    int t = r >> 3, b = r & 7;
    int tok = x[b * SEQ + t];
    const float4* src = (const float4*)(embed + (size_t)tok * HID);
    float4*       dst = (float4*)(emb + (size_t)r * HID);
    dst[threadIdx.x] = src[threadIdx.x];
}

// ---------------------------------------------------------------------------
// 2) One RNN timestep: hs[t] = tanh(emb_t @ Wih^T + hs[t-1] @ Whh^T + bih+bhh)
//    grid.x = 64, block = 32 (one wave per 16x16 tile). M padded 8->16 by row
//    duplication (upper half discarded). K-chunks of 32: 16 loads, 1 wait,
//    8 WMMAs into 4 rotating accumulators.
// ---------------------------------------------------------------------------
__device__ __forceinline__
void gemm_kchunks(const float* __restrict__ Arow, const float* __restrict__ Wcol,
                  int khalf, v8f acc[4]) {
    for (int kb = 0; kb < HID; kb += 32) {
        v2f a[8], b[8];
#pragma unroll
        for (int u = 0; u < 8; ++u) {
            int k = kb + 4 * u + 2 * khalf;
            a[u] = *(const v2f*)(Arow + k);
            b[u] = *(const v2f*)(Wcol + k);
        }
#pragma unroll
        for (int u = 0; u < 8; ++u)
            acc[u & 3] = wmma4(a[u], b[u], acc[u & 3]);
    }
}

__global__ __launch_bounds__(32)
void rnn_step_kernel(const float* __restrict__ emb,   // [T*8, H]
                     const float* __restrict__ Wih,   // [H, H]
                     const float* __restrict__ Whh,   // [H, H]
                     const float* __restrict__ bih,
                     const float* __restrict__ bhh,
                     float* __restrict__ hs,          // [T*8, H]
                     int t) {
    const int lane  = threadIdx.x;
    const int l15   = lane & 15;
    const int khalf = lane >> 4;
    const int col   = blockIdx.x * 16 + l15;

    v8f acc[4] = {v8f{}, v8f{}, v8f{}, v8f{}};

    const float* At = emb + (size_t)(t * BATCH + (l15 & 7)) * HID;
    gemm_kchunks(At, Wih + (size_t)col * HID, khalf, acc);

    if (t > 0) {
        const float* Ah = hs + (size_t)((t - 1) * BATCH + (l15 & 7)) * HID;
        gemm_kchunks(Ah, Whh + (size_t)col * HID, khalf, acc);
    }

    v8f tot = acc[0] + acc[1] + acc[2] + acc[3];
    float bsum = bih[col] + bhh[col];

    if (khalf == 0) {                 // C: vgpr j -> M=j (+8*khalf); keep M<8
        float* out = hs + (size_t)(t * BATCH) * HID;
#pragma unroll
        for (int j = 0; j < 8; ++j)
            out[(size_t)j * HID + col] = tanhf(tot[j] + bsum);
    }
}

// ---------------------------------------------------------------------------
// 3) Output projection: out[b][t][:] = hs[t*8+b] @ Wout^T + bout
//    Block = 256 thr (8 waves), block tile 64M x 128N, K-chunks of 32 double-
//    buffered in LDS via async-to-LDS. Wave w: M-tile (w>>1), N-tiles
//    4*(w&1)..+3 (A reused 4x from registers).
// ---------------------------------------------------------------------------
#define BM   64
#define BN   128
#define KC   32
#define LDST (KC + 2)    // padded row stride (floats), 8B-aligned, bank-spread

__global__ __launch_bounds__(256)
void proj_kernel(const float* __restrict__ hs,     // [2048, H]
                 const float* __restrict__ Wout,   // [V, H]
                 const float* __restrict__ bout,   // [V]
                 float* __restrict__ out) {        // [B, T, V]
    __shared__ float sA[2][BM][LDST];
    __shared__ float sB[2][BN][LDST];

    const int tid   = threadIdx.x;
    const int lane  = tid & 31;
    const int wv    = tid >> 5;          // 0..7
    const int l15   = lane & 15;
    const int khalf = lane >> 4;
    const int mt    = wv >> 1;           // wave's M-tile within block (0..3)
    const int ng    = (wv & 1) * 4;      // wave's first N-tile (0 or 4)

    const int cbase = blockIdx.x * BN;   // 250 blocks * 128 cols
    const int mbase = blockIdx.y * BM;   // 32 blocks * 64 rows

    const float* gA = hs   + (size_t)mbase * HID;
    const float* gB = Wout + (size_t)cbase * HID;

    // stage chunk `c` (k = c*KC) into buffer `buf`
    auto stage = [&](int c, int buf) {
        int kb = c * KC;
#pragma unroll
        for (int j = 0; j < 4; ++j) {            // A: 64 rows x 16 k-pairs
            int i  = tid + 256 * j;
            int r  = i >> 4;
            int kp = (i & 15) * 2;
            async_cp_b64(gA + (size_t)r * HID + kb + kp, &sA[buf][r][kp]);
        }
#pragma unroll
        for (int j = 0; j < 8; ++j) {            // B: 128 rows x 16 k-pairs
            int i  = tid + 256 * j;
            int r  = i >> 4;
            int kp = (i & 15) * 2;
            async_cp_b64(gB + (size_t)r * HID + kb + kp, &sB[buf][r][kp]);
        }
    };

    v8f acc[4] = {v8f{}, v8f{}, v8f{}, v8f{}};

    stage(0, 0);
    const int NC = HID / KC;                     // 32 chunks
    for (int c = 0; c < NC; ++c) {
        int buf = c & 1;
        async_wait_all();                        // this wave's stage(c) landed
        __syncthreads();                         // everyone's landed; prev buf free
        if (c + 1 < NC) stage(c + 1, buf ^ 1);
#pragma unroll
        for (int u = 0; u < KC / 4; ++u) {       // 8 WMMA k-steps
            int kk = 4 * u + 2 * khalf;
            v2f a = *(const v2f*)&sA[buf][16 * mt + l15][kk];
#pragma unroll
            for (int nt = 0; nt < 4; ++nt) {
                v2f b = *(const v2f*)&sB[buf][16 * (ng + nt) + l15][kk];
                acc[nt] = wmma4(a, b, acc[nt]);
            }
        }
    }

#pragma unroll
    for (int nt = 0; nt < 4; ++nt) {
        int col = cbase + 16 * (ng + nt) + l15;
        float bias = bout[col];
#pragma unroll
        for (int j = 0; j < 8; ++j) {
            int m = mbase + 16 * mt + j + 8 * khalf;   // hs row = t*8 + b
            int b = m & 7, t = m >> 3;
            out[((size_t)b * SEQ + t) * VOCAB + col] = acc[nt][j] + bias;
        }
    }
}

// ---------------------------------------------------------------------------
// 4) h_final = hs[T-1]
// ---------------------------------------------------------------------------
__global__ __launch_bounds__(256)
void hfinal_copy_kernel(const float* __restrict__ hs, float* __restrict__ dst) {
    int i = blockIdx.x * 256 + threadIdx.x;
    dst[i] = hs[(size_t)(SEQ - 1) * BATCH * HID + i];
}

// ---------------------------------------------------------------------------
extern "C" void kernel_launch(void* const* d_in, const int* in_sizes, int n_in,
                              void* d_out, int out_size, void* d_ws, size_t ws_size,
                              hipStream_t stream) {
    const int*   x     = (const int*)  d_in[0];
    const float* embed = (const float*)d_in[1];
    const float* Wih   = (const float*)d_in[2];
    const float* bih   = (const float*)d_in[3];
    const float* Whh   = (const float*)d_in[4];
    const float* bhh   = (const float*)d_in[5];
    const float* Wout  = (const float*)d_in[6];
    const float* bout  = (const float*)d_in[7];
    float* out = (float*)d_out;

    float* emb = (float*)d_ws;                                   // [T*8, H] 8 MB
    float* hs  = emb + (size_t)SEQ * BATCH * HID;                // [T*8, H] 8 MB

    emb_gather_kernel<<<SEQ * BATCH, 256, 0, stream>>>(x, embed, emb);

    for (int t = 0; t < SEQ; ++t)
        rnn_step_kernel<<<HID / 16, 32, 0, stream>>>(emb, Wih, Whh, bih, bhh, hs, t);

    proj_kernel<<<dim3(VOCAB / BN, (SEQ * BATCH) / BM), 256, 0, stream>>>(hs, Wout, bout, out);

    hfinal_copy_kernel<<<(BATCH * HID) / 256, 256, 0, stream>>>(
        hs, out + (size_t)BATCH * SEQ * VOCAB);
}